// PDConv_11467562680752
// MI455X (gfx1250) — compile-verified
//
#include <hip/hip_runtime.h>
#include <hip/hip_bf16.h>
#include <stdint.h>

// ---------------------------------------------------------------------------
// PDConv for gfx1250 (MI455X).
// pooled[b,c,h,w] = max(0, max over even-coord taps with mask bit set)
// out = concat([pooled, x_pre], channel axis)
// Bandwidth-bound (~54 MiB total traffic -> ~2.4us at 23.3 TB/s).
// CDNA5 path: GLOBAL_LOAD_ASYNC_TO_LDS_B128 staging + s_wait_asynccnt,
// b128 vector loads for the mask planes, b64/b128 stores.
// ---------------------------------------------------------------------------

#if defined(__AMDGCN__) && __has_builtin(__builtin_amdgcn_global_load_async_to_lds_b128)
#define PD_ASYNC_LDS 1
#else
#define PD_ASYNC_LDS 0
#endif

#if PD_ASYNC_LDS
typedef int pd_v4i __attribute__((ext_vector_type(4)));
typedef __attribute__((address_space(1))) pd_v4i* pd_gv4i_p;  // global
typedef __attribute__((address_space(3))) pd_v4i* pd_lv4i_p;  // LDS
#endif

__device__ __forceinline__ void pd_async_copy16(const float* gp, float* lp) {
#if PD_ASYNC_LDS
  __builtin_amdgcn_global_load_async_to_lds_b128(
      (pd_gv4i_p)gp, (pd_lv4i_p)lp, /*offset=*/0, /*cpol=*/0);
#else
  *(float4*)lp = *(const float4*)gp;
#endif
}

__device__ __forceinline__ void pd_async_wait0() {
#if PD_ASYNC_LDS
#if __has_builtin(__builtin_amdgcn_s_wait_asynccnt)
  __builtin_amdgcn_s_wait_asynccnt(0);
#else
  asm volatile("s_wait_asynccnt 0" ::: "memory");
#endif
#endif
}

// ---------------------------------------------------------------------------
// pool_kernel: grid = (8 h-tiles, 8 channel-groups, 4 batches), 256 threads.
// Each block: 16 output rows x 128 cols x 8 channels.
// ---------------------------------------------------------------------------
__global__ __launch_bounds__(256) void pdconv_pool_kernel(
    const float* __restrict__ x,      // (4,64,64,64)
    const int*   __restrict__ gm,     // (4,9,128,128)
    float*       __restrict__ out) {  // (4,128,128,128); we fill ch 0..63
  const int tid   = threadIdx.x;
  const int tileH = blockIdx.x;   // 0..7  -> rows h0..h0+15
  const int cg    = blockIdx.y;   // 0..7  -> channels c0..c0+7
  const int b     = blockIdx.z;   // 0..3
  const int h0    = tileH * 16;
  const int hsb   = h0 >> 1;      // source-row base in x
  const int c0    = cg * 8;

  __shared__ unsigned int bits[16 * 128];   // 9-bit mask bitmap per pixel (8 KiB)
  __shared__ float xt[8][9][68];            // 8 ch x 9 src rows x 64(+pad) cols

  // --- Stage A: build mask bitmaps with b128 loads (4 pixels x 9 planes) ---
  {
    const int4* gm4 = (const int4*)gm;
    const int gbase4 = (b * 9 * 16384 + h0 * 128) >> 2;  // int4 index
#pragma unroll
    for (int k = 0; k < 2; ++k) {
      const int p4 = tid + k * 256;       // int4 group: 0..511 (2048 px / 4)
      unsigned int m0 = 0, m1 = 0, m2 = 0, m3 = 0;
#pragma unroll
      for (int kk = 0; kk < 9; ++kk) {
        const int4 v = gm4[gbase4 + kk * 4096 + p4];
        m0 |= (v.x != 0 ? 1u : 0u) << kk;
        m1 |= (v.y != 0 ? 1u : 0u) << kk;
        m2 |= (v.z != 0 ? 1u : 0u) << kk;
        m3 |= (v.w != 0 ? 1u : 0u) << kk;
      }
      *(uint4*)&bits[p4 * 4] = make_uint4(m0, m1, m2, m3);
    }
  }

  // --- Stage B: zero the LDS tile padding (edge col 64..67, OOB rows) ---
  if (tid < 72) {
    const int ch  = tid / 9;
    const int row = tid - ch * 9;
    float4* rp = (float4*)&xt[ch][row][0];  // 68 floats = 17 float4, 16B aligned
    rp[16] = make_float4(0.f, 0.f, 0.f, 0.f);
    if (hsb + row >= 64) {
#pragma unroll
      for (int s = 0; s < 16; ++s) rp[s] = make_float4(0.f, 0.f, 0.f, 0.f);
    }
  }

  // --- Stage C: async-stage x tile: 8 ch x 9 rows x 64 cols = 1152 b128 ---
  for (int id = tid; id < 1152; id += 256) {
    const int ch  = id / 144;
    const int rid = id - ch * 144;
    const int row = rid >> 4;
    const int seg = rid & 15;
    const int hs  = hsb + row;
    if (hs < 64) {
      const float* gp = x + ((((b * 64 + c0 + ch) * 64 + hs) * 64) + seg * 4);
      pd_async_copy16(gp, &xt[ch][row][seg * 4]);
    }
  }
  pd_async_wait0();
  __syncthreads();

  // --- Stage D: compute 2x2 output quads, branch-free via bit selects ---
  const size_t outBase = ((size_t)(b * 128 + c0) * 128 + h0) * 128;
  for (int q = tid; q < 512; q += 256) {
    const int qr = q >> 6;                  // 0..7 quad-row
    const int qc = q & 63;                  // 0..63 quad-col
    const int r0 = qr * 2;
    const unsigned int b00 = bits[r0 * 128 + qc * 2];
    const unsigned int b01 = bits[r0 * 128 + qc * 2 + 1];
    const unsigned int b10 = bits[r0 * 128 + 128 + qc * 2];
    const unsigned int b11 = bits[r0 * 128 + 128 + qc * 2 + 1];
#pragma unroll
    for (int ch = 0; ch < 8; ++ch) {
      const float v00 = xt[ch][qr][qc];
      const float v01 = xt[ch][qr][qc + 1];
      const float v10 = xt[ch][qr + 1][qc];
      const float v11 = xt[ch][qr + 1][qc + 1];

      // (h even, w even): tap k=4 -> v00
      const float o00 = fmaxf(0.f, (b00 & (1u << 4)) ? v00 : 0.f);
      // (h even, w odd): k=3 -> v00, k=5 -> v01
      const float o01 = fmaxf(0.f, fmaxf((b01 & (1u << 3)) ? v00 : 0.f,
                                         (b01 & (1u << 5)) ? v01 : 0.f));
      // (h odd, w even): k=1 -> v00, k=7 -> v10
      const float o10 = fmaxf(0.f, fmaxf((b10 & (1u << 1)) ? v00 : 0.f,
                                         (b10 & (1u << 7)) ? v10 : 0.f));
      // (h odd, w odd): k=0 -> v00, k=2 -> v01, k=6 -> v10, k=8 -> v11
      float o11 = fmaxf(fmaxf((b11 & (1u << 0)) ? v00 : 0.f,
                              (b11 & (1u << 2)) ? v01 : 0.f),
                        fmaxf((b11 & (1u << 6)) ? v10 : 0.f,
                              (b11 & (1u << 8)) ? v11 : 0.f));
      o11 = fmaxf(o11, 0.f);

      float* op = out + outBase + (size_t)ch * 16384 + (size_t)r0 * 128 + qc * 2;
      *(float2*)(op)       = make_float2(o00, o01);
      *(float2*)(op + 128) = make_float2(o10, o11);
    }
  }
}

// ---------------------------------------------------------------------------
// copy kernel: x_pre -> out channels 64..127 (contiguous per batch), float4.
// ---------------------------------------------------------------------------
__global__ __launch_bounds__(256) void pdconv_copy_kernel(
    const float4* __restrict__ src, float4* __restrict__ dst) {
  const size_t i = (size_t)blockIdx.x * 256 + threadIdx.x;  // 0..1048575
  const int per_b = 64 * 4096;                              // float4 per batch
  const int b = (int)(i / per_b);
  const int rem = (int)(i - (size_t)b * per_b);
  dst[(size_t)(b * 128 + 64) * 4096 + rem] = src[i];
}

extern "C" void kernel_launch(void* const* d_in, const int* in_sizes, int n_in,
                              void* d_out, int out_size, void* d_ws, size_t ws_size,
                              hipStream_t stream) {
  const float* x_pre = (const float*)d_in[0];  // (4,64,128,128)
  const float* x     = (const float*)d_in[1];  // (4,64,64,64)
  const int*   gm    = (const int*)d_in[2];    // (4,9,128,128)
  float* out = (float*)d_out;                  // (4,128,128,128)

  dim3 pgrid(8, 8, 4);
  hipLaunchKernelGGL(pdconv_pool_kernel, pgrid, dim3(256), 0, stream, x, gm, out);

  const int n4 = (4 * 64 * 16384) / 4;         // 1,048,576 float4
  hipLaunchKernelGGL(pdconv_copy_kernel, dim3(n4 / 256), dim3(256), 0, stream,
                     (const float4*)x_pre, (float4*)out);
}